// Model_72696616452468
// MI455X (gfx1250) — compile-verified
//
#include <hip/hip_runtime.h>
#include <hip/hip_bf16.h>

typedef float v2f __attribute__((ext_vector_type(2)));
typedef float v8f __attribute__((ext_vector_type(8)));

#define WMMA_F32X4 __builtin_amdgcn_wmma_f32_16x16x4_f32

static constexpr int N_NODES = 50000;
static constexpr int N_EDGES = 800000;
static constexpr int IN_CH   = 128;
static constexpr int HID     = 32;
static constexpr int OUT_CH  = 10;
static constexpr int N_GRAPH = 256;
static constexpr int MAX_ITER = 8;
static constexpr int NEUMANN_K = 10;
// ALPHA = 1.0 -> c = 0.5, 1/(1+ALPHA) = 0.5
static constexpr float NEUMANN_C = 0.5f;

// ---------------------------------------------------------------------------
// 32x32 Cayley weight: S = B - B^T; solve (I+S)Q = (I-S); WT[k][h] = 0.5*Q[h][k]
// One block, 256 threads, Gauss-Jordan on a 32x64 augmented system in LDS.
// ---------------------------------------------------------------------------
__global__ void cayley_kernel(const float* __restrict__ B, float* __restrict__ WT) {
  __shared__ float M[32][64];
  const int tid = threadIdx.x;          // 256 threads
  const int r   = tid >> 3;             // row 0..31
  const int c0  = (tid & 7) * 8;        // 8 contiguous columns

#pragma unroll
  for (int j = 0; j < 8; ++j) {
    int c = c0 + j;
    if (c < 32) {
      float s = B[r * 32 + c] - B[c * 32 + r];
      M[r][c] = (r == c ? 1.0f : 0.0f) + s;      // I + S
    } else {
      int cc = c - 32;
      float s = B[r * 32 + cc] - B[cc * 32 + r];
      M[r][c] = (r == cc ? 1.0f : 0.0f) - s;     // I - S (RHS)
    }
  }
  __syncthreads();

  for (int k = 0; k < 32; ++k) {
    float piv = M[k][k];
    __syncthreads();
    if (r == k) {
      float inv = 1.0f / piv;
#pragma unroll
      for (int j = 0; j < 8; ++j) M[k][c0 + j] *= inv;
    }
    __syncthreads();
    float f = M[r][k];
    __syncthreads();
    if (r != k) {
#pragma unroll
      for (int j = 0; j < 8; ++j) M[r][c0 + j] -= f * M[k][c0 + j];
    }
    __syncthreads();
  }

  // Q[h][k] = M[h][32+k];  WT stored (K=32) x 32 row-major: WT[k*32+h] = 0.5*Q[h][k]
#pragma unroll
  for (int j = 0; j < 8; ++j) {
    int c = c0 + j;
    if (c >= 32) {
      int k = c - 32;
      WT[k * 32 + r] = 0.5f * M[r][c];
    }
  }
}

// ---------------------------------------------------------------------------
// C(N x 32) = A(N x K) @ B(K x 32) [+ bias32], fp32 via V_WMMA_F32_16X16X4_F32.
// A node-major (row stride K), C node-major N x 32. B fragments are PRE-PACKED
// into LDS in the exact per-lane v2f layout the WMMA B-operand wants:
//   sBp[(s*2 + t)*32 + lane] = { B[4s+2*(lane/16)  ][16t + lane%16],
//                                B[4s+2*(lane/16)+1][16t + lane%16] }
// so each fragment is ONE aligned ds_load_b64 into an even VGPR pair (no movs).
// One wave computes a 16-node x 32-channel tile. N multiple of 16 (50000=3125*16).
// ---------------------------------------------------------------------------
__global__ void gemm_n32(const float* __restrict__ A, const float* __restrict__ B,
                         const float* __restrict__ bias32, float* __restrict__ C,
                         int n, int K) {
  __shared__ v2f sBp[(IN_CH / 4) * 2 * 32];   // max K=128 -> 2048 v2f = 16 KB
  const int ksteps = K >> 2;

  for (int i = threadIdx.x; i < ksteps * 2 * 32; i += blockDim.x) {
    const int l  = i & 31;          // destination lane
    const int t  = (i >> 5) & 1;    // h-tile
    const int s  = i >> 6;          // k-step
    const int k0 = 4 * s + 2 * (l >> 4);
    const int cc = 16 * t + (l & 15);
    v2f v;
    v.x = B[k0 * 32 + cc];
    v.y = B[(k0 + 1) * 32 + cc];
    sBp[i] = v;
  }
  __syncthreads();

  const int lane = threadIdx.x & 31;
  const int wave = threadIdx.x >> 5;
  const int wpb  = blockDim.x >> 5;
  const int m    = lane & 15;        // A row within tile / D column within tile
  const int kh   = lane >> 4;        // K-half select (0: K+0/1, 1: K+2/3)
  const int ntiles = n >> 4;

  for (int tile = blockIdx.x * wpb + wave; tile < ntiles; tile += gridDim.x * wpb) {
    v8f c0 = {};
    v8f c1 = {};
    const float* arow = A + (size_t)(tile * 16 + m) * K;

    for (int s = 0; s < ksteps; ++s) {
      const int k0 = 4 * s + 2 * kh;
      const v2f a  = *(const v2f*)(arow + k0);       // 8B-aligned global b64 load
      const v2f b0 = sBp[(s * 2 + 0) * 32 + lane];   // single ds_load_b64
      const v2f b1 = sBp[(s * 2 + 1) * 32 + lane];   // single ds_load_b64
      c0 = WMMA_F32X4(false, a, false, b0, (short)0, c0, false, false);
      c1 = WMMA_F32X4(false, a, false, b1, (short)0, c1, false, false);
    }

    const float bb0 = bias32 ? bias32[m]      : 0.0f;
    const float bb1 = bias32 ? bias32[16 + m] : 0.0f;
    const int nbase = tile * 16;
#pragma unroll
    for (int r = 0; r < 8; ++r) {
      const int node = nbase + r + 8 * kh;     // D layout: row = r + 8*(lane/16)
      C[(size_t)node * 32 + m]      = c0[r] + bb0;
      C[(size_t)node * 32 + 16 + m] = c1[r] + bb1;
    }
  }
}

// ---------------------------------------------------------------------------
// CSC build: histogram -> exclusive scan -> scatter
// ---------------------------------------------------------------------------
__global__ void zero_i32(int* __restrict__ p, int n) {
  int i = blockIdx.x * blockDim.x + threadIdx.x;
  if (i < n) p[i] = 0;
}
__global__ void zero_f32(float* __restrict__ p, int n) {
  int i = blockIdx.x * blockDim.x + threadIdx.x;
  if (i < n) p[i] = 0.0f;
}

__global__ void hist_kernel(const int* __restrict__ col, int* __restrict__ cnt, int e) {
  int i = blockIdx.x * blockDim.x + threadIdx.x;
  if (i < e) atomicAdd(&cnt[col[i]], 1);
}

// single-block chunked Hillis-Steele exclusive scan: cnt(n) -> starts(n+1)
__global__ void scan_kernel(const int* __restrict__ cnt, int* __restrict__ starts, int n) {
  __shared__ int buf[1024];
  __shared__ int carry_s;
  const int tid = threadIdx.x;
  if (tid == 0) carry_s = 0;
  __syncthreads();
  for (int base = 0; base < n; base += 1024) {
    const int i = base + tid;
    const int v = (i < n) ? cnt[i] : 0;
    buf[tid] = v;
    __syncthreads();
    for (int off = 1; off < 1024; off <<= 1) {
      int t = (tid >= off) ? buf[tid - off] : 0;
      __syncthreads();
      buf[tid] += t;
      __syncthreads();
    }
    const int incl  = buf[tid];
    const int carry = carry_s;
    if (i < n) starts[i] = carry + incl - v;
    __syncthreads();
    if (tid == 1023) carry_s = carry + incl;
    __syncthreads();
  }
  if (tid == 0) starts[n] = carry_s;
}

__global__ void scatter_build(const int* __restrict__ row, const int* __restrict__ col,
                              const float* __restrict__ ew, const int* __restrict__ starts,
                              int* __restrict__ cur, int* __restrict__ srow,
                              float* __restrict__ sw, int e) {
  int i = blockIdx.x * blockDim.x + threadIdx.x;
  if (i < e) {
    const int c  = col[i];
    const int pos = starts[c] + atomicAdd(&cur[c], 1);
    srow[pos] = row[i];
    sw[pos]   = ew[i];
  }
}

// ---------------------------------------------------------------------------
// SpMM gather: one wave per node, lane = channel; atomic-free, L2-resident.
// term[n][:] = scale * sum_e w_e * tmp[row_e][:];  acc += term.
// Edge loop unrolled x2 for memory-level parallelism against L2 latency.
// ---------------------------------------------------------------------------
__global__ void spmm_gather(const float* __restrict__ tmp, const int* __restrict__ starts,
                            const int* __restrict__ srow, const float* __restrict__ sw,
                            float scale, float* __restrict__ term, float* __restrict__ acc,
                            int n) {
  const int node = blockIdx.x * (blockDim.x >> 5) + (threadIdx.x >> 5);
  const int h    = threadIdx.x & 31;
  if (node >= n) return;
  const int s = starts[node];
  const int e = starts[node + 1];
  float a0 = 0.0f, a1 = 0.0f;
  int i = s;
  for (; i + 1 < e; i += 2) {
    const int   r0 = srow[i];
    const int   r1 = srow[i + 1];
    const float w0 = sw[i];
    const float w1 = sw[i + 1];
    a0 += w0 * tmp[(size_t)r0 * 32 + h];   // two independent 128B gathers in flight
    a1 += w1 * tmp[(size_t)r1 * 32 + h];
  }
  if (i < e) a0 += sw[i] * tmp[(size_t)srow[i] * 32 + h];
  const float a = (a0 + a1) * scale;
  const size_t o = (size_t)node * 32 + h;
  term[o] = a;
  acc[o] += a;
}

// ---------------------------------------------------------------------------
// PR elementwise steps (ALPHA = 1): u12 = 2z-u; v = u12 + bias; term=acc=v
// post: u_new = acc - u12; z_new = relu(u_new)
// ---------------------------------------------------------------------------
__global__ void pre_pr(const float* __restrict__ z, const float* __restrict__ u,
                       const float* __restrict__ bias, float* __restrict__ u12,
                       float* __restrict__ term, float* __restrict__ acc, int n) {
  int i = blockIdx.x * blockDim.x + threadIdx.x;
  if (i < n) {
    const float a = 2.0f * z[i] - u[i];
    u12[i] = a;
    const float v = a + bias[i];
    term[i] = v;
    acc[i]  = v;
  }
}

__global__ void post_pr(const float* __restrict__ acc, const float* __restrict__ u12,
                        float* __restrict__ u, float* __restrict__ z, int n) {
  int i = blockIdx.x * blockDim.x + threadIdx.x;
  if (i < n) {
    const float un = acc[i] - u12[i];   // 2*(acc*0.5) - u12
    u[i] = un;
    z[i] = fmaxf(un, 0.0f);
  }
}

// ---------------------------------------------------------------------------
// Pooling (done once; atomics into 256x32 f32) + decoder
// ---------------------------------------------------------------------------
__global__ void pool_kernel(const float* __restrict__ z, const int* __restrict__ batch,
                            float* __restrict__ pooled, int nnodes) {
  int t = blockIdx.x * blockDim.x + threadIdx.x;
  const int node = t >> 5;
  const int h    = t & 31;
  if (node < nnodes) {
    const float v = fmaxf(z[(size_t)node * 32 + h], 0.0f);
    atomicAdd(&pooled[batch[node] * 32 + h], v);
  }
}

__global__ void out_gemm(const float* __restrict__ pooled, const float* __restrict__ dec_w,
                         float* __restrict__ out) {
  int t = blockIdx.x * blockDim.x + threadIdx.x;
  if (t < N_GRAPH * OUT_CH) {
    const int g = t / OUT_CH;
    const int o = t % OUT_CH;
    float s = 0.0f;
#pragma unroll
    for (int h = 0; h < HID; ++h) s += pooled[g * 32 + h] * dec_w[h * OUT_CH + o];
    out[t] = s;
  }
}

// ---------------------------------------------------------------------------
extern "C" void kernel_launch(void* const* d_in, const int* in_sizes, int n_in,
                              void* d_out, int out_size, void* d_ws, size_t ws_size,
                              hipStream_t stream) {
  (void)in_sizes; (void)n_in; (void)out_size; (void)ws_size;

  const float* x       = (const float*)d_in[0];
  const int*   eidx    = (const int*)  d_in[1];   // [0..E) = row, [E..2E) = col
  const float* ew      = (const float*)d_in[2];
  /* d_in[3] = num_nodes scalar (compile-time constant here) */
  const int*   batch   = (const int*)  d_in[4];
  const float* enc_w   = (const float*)d_in[5];
  const float* enc_b   = (const float*)d_in[6];
  const float* dec_w   = (const float*)d_in[7];
  const float* cayleyB = (const float*)d_in[8];
  float* out = (float*)d_out;

  const int* row = eidx;
  const int* col = eidx + N_EDGES;

  // workspace carve-out (256B aligned)
  char* w = (char*)d_ws;
  auto carve = [&](size_t bytes) -> char* {
    char* p = w;
    w += (bytes + 255) & ~(size_t)255;
    return p;
  };
  const size_t NV = (size_t)N_NODES * 32;
  float* WT     = (float*)carve(32 * 32 * sizeof(float));
  float* bias   = (float*)carve(NV * sizeof(float));
  float* z      = (float*)carve(NV * sizeof(float));
  float* u      = (float*)carve(NV * sizeof(float));
  float* u12    = (float*)carve(NV * sizeof(float));
  float* term   = (float*)carve(NV * sizeof(float));
  float* acc    = (float*)carve(NV * sizeof(float));
  float* tmp    = (float*)carve(NV * sizeof(float));
  int*   starts = (int*)  carve((N_NODES + 1) * sizeof(int));
  int*   cnt    = (int*)  carve(N_NODES * sizeof(int)); // histogram, then cursor
  int*   srow   = (int*)  carve(N_EDGES * sizeof(int));
  float* sw     = (float*)carve(N_EDGES * sizeof(float));
  float* pooled = (float*)carve(N_GRAPH * 32 * sizeof(float));

  const int gEdges = (N_EDGES + 255) / 256;            // 3125
  const int gNV    = (int)((NV + 255) / 256);          // 6250
  const int gGemm  = ((N_NODES / 16) + 7) / 8;         // 391 blocks x 8 waves
  const int gNodes = (N_NODES + 7) / 8;                // wave-per-node kernels

  // 1) Cayley weight (W^T, scaled by 1-m)
  cayley_kernel<<<1, 256, 0, stream>>>(cayleyB, WT);

  // 2) encoder: bias = x @ enc_w + enc_b  (node-major N x 32)
  gemm_n32<<<gGemm, 256, 0, stream>>>(x, enc_w, enc_b, bias, N_NODES, IN_CH);

  // 3) z = u = 0
  zero_f32<<<gNV, 256, 0, stream>>>(z, (int)NV);
  zero_f32<<<gNV, 256, 0, stream>>>(u, (int)NV);

  // 4) CSC build (once per call; reused by all 80 SpMM passes)
  zero_i32<<<(N_NODES + 255) / 256, 256, 0, stream>>>(cnt, N_NODES);
  hist_kernel<<<gEdges, 256, 0, stream>>>(col, cnt, N_EDGES);
  scan_kernel<<<1, 1024, 0, stream>>>(cnt, starts, N_NODES);
  zero_i32<<<(N_NODES + 255) / 256, 256, 0, stream>>>(cnt, N_NODES);
  scatter_build<<<gEdges, 256, 0, stream>>>(row, col, ew, starts, cnt, srow, sw, N_EDGES);

  // 5) Peaceman-Rachford iterations with Neumann-10 inverse
  for (int it = 0; it < MAX_ITER; ++it) {
    pre_pr<<<gNV, 256, 0, stream>>>(z, u, bias, u12, term, acc, (int)NV);
    for (int k = 0; k < NEUMANN_K; ++k) {
      gemm_n32<<<gGemm, 256, 0, stream>>>(term, WT, nullptr, tmp, N_NODES, HID);
      spmm_gather<<<gNodes, 256, 0, stream>>>(tmp, starts, srow, sw, NEUMANN_C,
                                              term, acc, N_NODES);
    }
    post_pr<<<gNV, 256, 0, stream>>>(acc, u12, u, z, (int)NV);
  }

  // 6) pooling + decoder
  zero_f32<<<(N_GRAPH * 32 + 255) / 256, 256, 0, stream>>>(pooled, N_GRAPH * 32);
  pool_kernel<<<gNodes, 256, 0, stream>>>(z, batch, pooled, N_NODES);
  out_gemm<<<(N_GRAPH * OUT_CH + 255) / 256, 256, 0, stream>>>(pooled, dec_w, out);
}